// HashFlagEmbedding_11553462026942
// MI455X (gfx1250) — compile-verified
//
#include <hip/hip_runtime.h>

#ifndef __has_builtin
#define __has_builtin(x) 0
#endif

typedef unsigned long long u64;
typedef int v4i __attribute__((ext_vector_type(4)));

#define GLOBAL_AS __attribute__((address_space(1)))
#define LDS_AS    __attribute__((address_space(3)))

#define BLK 256
#define FLAG_DIM 64

__device__ __forceinline__ u64 rotr64(u64 x, unsigned n) {
    return (x >> n) | (x << (64u - n));
}

__device__ __forceinline__ float tanh_fast(float x) {
#if __has_builtin(__builtin_amdgcn_tanhf)
    return __builtin_amdgcn_tanhf(x);                 // v_tanh_f32 (gfx1250)
#elif __has_builtin(__builtin_amdgcn_tanh_f32)
    return __builtin_amdgcn_tanh_f32(x);
#else
    float e = __expf(2.0f * x);
    return (e - 1.0f) / (e + 1.0f);
#endif
}

__global__ __launch_bounds__(BLK) void blake2b_embed_kernel(
    const long long* __restrict__ ids, float* __restrict__ out, int N)
{
    __shared__ __align__(16) float smem[BLK * FLAG_DIM];   // 64 KB: block tile, global layout

    const int tid = threadIdx.x;
    const int gid = blockIdx.x * BLK + tid;
    // All threads compute; tail lanes redo id N-1 (result never stored).
    const int cgid = gid < N ? gid : (N - 1);

    const u64 n = (u64)ids[cgid];

    // ---- decimal string -> message words (<=20 bytes -> m0..m2) ----
    int nd = 1;
    { u64 t = n; while (t >= 10ull) { t /= 10ull; ++nd; } }   // const-div by 10
    u64 m0 = 0, m1 = 0, m2 = 0;
    { u64 t = n;
      for (int k = 0; k < nd; ++k) {
          u64 d = t % 10ull; t /= 10ull;                      // const-div by 10
          int i = nd - 1 - k;                                 // MS digit first
          u64 b = d + 48ull;
          if (i < 8)        m0 |= b << (8 * i);
          else if (i < 16)  m1 |= b << (8 * (i - 8));
          else              m2 |= b << (8 * (i - 16));
      }
    }

    // ---- BLAKE2b single-block compression, digest_size=64 ----
    const u64 IV[8] = {
        0x6a09e667f3bcc908ull, 0xbb67ae8584caa73bull,
        0x3c6ef372fe94f82bull, 0xa54ff53a5f1d36f1ull,
        0x510e527fade682d1ull, 0x9b05688c2b3e6c1full,
        0x1f83d9abfb41bd6bull, 0x5be0cd19137e2179ull };
    static constexpr unsigned char S[12][16] = {
        {0,1,2,3,4,5,6,7,8,9,10,11,12,13,14,15},
        {14,10,4,8,9,15,13,6,1,12,0,2,11,7,5,3},
        {11,8,12,0,5,2,15,13,10,14,3,6,7,1,9,4},
        {7,9,3,1,13,12,11,14,2,6,5,10,4,0,15,8},
        {9,0,5,7,2,4,10,15,14,1,11,12,6,8,3,13},
        {2,12,6,10,0,11,8,3,4,13,7,5,15,14,1,9},
        {12,5,1,15,14,13,4,10,0,7,6,3,9,2,8,11},
        {13,11,7,14,12,1,3,9,5,0,15,4,8,6,2,10},
        {6,15,14,9,11,3,0,8,12,2,13,7,1,4,10,5},
        {10,2,8,4,7,6,1,5,15,11,9,14,3,12,13,0},
        {0,1,2,3,4,5,6,7,8,9,10,11,12,13,14,15},
        {14,10,4,8,9,15,13,6,1,12,0,2,11,7,5,3} };

    const u64 m[16] = { m0, m1, m2, 0,0,0,0,0,0,0,0,0,0,0,0,0 };

    const u64 h0 = IV[0] ^ 0x01010040ull;   // 0x01010000 ^ digest_size(64)
    u64 v[16];
    v[0] = h0;
    #pragma unroll
    for (int i = 1; i < 8; ++i) v[i] = IV[i];
    #pragma unroll
    for (int i = 0; i < 8; ++i) v[8 + i] = IV[i];
    v[12] ^= (u64)(unsigned)nd;             // t0 = message byte length
    v[14] ^= 0xFFFFFFFFFFFFFFFFull;         // final block

#define G_MIX(a,b,c,d,x,y)                                        \
    v[a] = v[a] + v[b] + (x); v[d] = rotr64(v[d] ^ v[a], 32); \
    v[c] = v[c] + v[d];       v[b] = rotr64(v[b] ^ v[c], 24); \
    v[a] = v[a] + v[b] + (y); v[d] = rotr64(v[d] ^ v[a], 16); \
    v[c] = v[c] + v[d];       v[b] = rotr64(v[b] ^ v[c], 63);

    #pragma unroll
    for (int r = 0; r < 12; ++r) {
        G_MIX(0,4, 8,12, m[S[r][ 0]], m[S[r][ 1]]);
        G_MIX(1,5, 9,13, m[S[r][ 2]], m[S[r][ 3]]);
        G_MIX(2,6,10,14, m[S[r][ 4]], m[S[r][ 5]]);
        G_MIX(3,7,11,15, m[S[r][ 6]], m[S[r][ 7]]);
        G_MIX(0,5,10,15, m[S[r][ 8]], m[S[r][ 9]]);
        G_MIX(1,6,11,12, m[S[r][10]], m[S[r][11]]);
        G_MIX(2,7, 8,13, m[S[r][12]], m[S[r][13]]);
        G_MIX(3,4, 9,14, m[S[r][14]], m[S[r][15]]);
    }
#undef G_MIX

    // ---- digest bytes -> scale -> tanh -> accumulate norm ----
    float f[FLAG_DIM];
    float sum = 0.0f;
    #pragma unroll
    for (int i = 0; i < 8; ++i) {
        const u64 w = (i == 0 ? h0 : IV[i]) ^ v[i] ^ v[i + 8];
        #pragma unroll
        for (int b = 0; b < 8; ++b) {
            float x = (float)(unsigned)((w >> (8 * b)) & 0xFFull);  // v_cvt_f32_ubyteN
            x = x * (1.0f / 127.5f) - 1.0f;
            x = tanh_fast(1.5f * x);
            f[i * 8 + b] = x;
            sum += x * x;
        }
    }
    const float inv = 1.0f / (__builtin_sqrtf(sum) + 1e-6f);

    // ---- stage normalized vector into LDS in the exact global layout ----
    {
        float4* sv = (float4*)&smem[tid * FLAG_DIM];
        #pragma unroll
        for (int k = 0; k < FLAG_DIM / 4; ++k) {
            float4 val;
            val.x = f[4 * k + 0] * inv;
            val.y = f[4 * k + 1] * inv;
            val.z = f[4 * k + 2] * inv;
            val.w = f[4 * k + 3] * inv;
            sv[k] = val;                                   // ds_store_b128
        }
    }
    __syncthreads();

    // ---- coalesced async LDS -> global streaming stores (ASYNCcnt path) ----
    // Non-temporal hint: 256 MB write-once stream > 192 MB L2.
    const long long blockBase = (long long)blockIdx.x * (BLK * FLAG_DIM);
    const long long total = (long long)N * FLAG_DIM;
    LDS_AS float* sbase = (LDS_AS float*)smem;
    #pragma unroll
    for (int k = 0; k < FLAG_DIM / 4; ++k) {            // 16 iterations
        const int q = k * BLK + tid;                    // float4 index in block tile
        const long long fidx = blockBase + (long long)q * 4;
        if (fidx < total) {
            float* gp = out + fidx;
#if __has_builtin(__builtin_amdgcn_global_store_async_from_lds_b128)
            __builtin_amdgcn_global_store_async_from_lds_b128(
                (GLOBAL_AS v4i*)gp, (LDS_AS v4i*)(sbase + (unsigned)q * 4),
                0, /*cpol: TH_STORE_NT*/ 1);
#else
            const u64 gaddr = (u64)gp;
            const unsigned laddr = (unsigned)(size_t)(sbase + (unsigned)q * 4);
            asm volatile("global_store_async_from_lds_b128 %0, %1, off th:TH_STORE_NT"
                         :: "v"(gaddr), "v"(laddr) : "memory");
#endif
        }
    }

#if __has_builtin(__builtin_amdgcn_s_wait_asynccnt)
    __builtin_amdgcn_s_wait_asynccnt(0);
#else
    asm volatile("s_wait_asynccnt 0" ::: "memory");
#endif
}

extern "C" void kernel_launch(void* const* d_in, const int* in_sizes, int n_in,
                              void* d_out, int out_size, void* d_ws, size_t ws_size,
                              hipStream_t stream) {
    const long long* ids = (const long long*)d_in[0];   // reference dtype: int64
    float* out = (float*)d_out;                         // float32 output, N x 64
    const int N = in_sizes[0];
    const int blocks = (N + BLK - 1) / BLK;
    hipLaunchKernelGGL(blake2b_embed_kernel, dim3(blocks), dim3(BLK), 0, stream,
                       ids, out, N);
}